// BandSALayer3D_26018911879482
// MI455X (gfx1250) — compile-verified
//
#include <hip/hip_runtime.h>
#include <hip/hip_bf16.h>
#include <math.h>

typedef float v2f __attribute__((ext_vector_type(2)));
typedef float v8f __attribute__((ext_vector_type(8)));

namespace {
constexpr int kN = 2;
constexpr int kF = 64;        // features
constexpr int kC = 31;        // bands
constexpr int kHW = 128 * 128;
constexpr int kHeads = 8;
constexpr int kP = 4;                       // pixels per tile
constexpr int kIters = 8;                   // pixel tiles per workgroup
constexpr int kPixPerBlock = kP * kIters;   // 32
constexpr int kXLD = 132;                   // X/V/Val LDS row stride (124 cols + pad)
constexpr int kVWLD = 65;                   // v_w LDS stride (odd -> no bank clash)
constexpr int kCWLD = 129;                  // conv_w LDS stride
constexpr int kATLD = 33;                   // attn LDS stride
}

// ---- f32 WMMA fragment helpers (16x16x4, wave32) ----
// A 16x4: lane%16 = M row, VGPR j with lane-half u holds K = 2u + j
__device__ __forceinline__ v2f frag_a(const float* base, int ld, int row0, int k0, int lane) {
  const int r = row0 + (lane & 15);
  const int kk = k0 + 2 * (lane >> 4);
  v2f a;
  a.x = base[r * ld + kk];
  a.y = base[r * ld + kk + 1];
  return a;
}
// B 4x16: lane%16 = N col, VGPR j with lane-half u holds K = 2u + j
__device__ __forceinline__ v2f frag_b(const float* base, int ld, int k0, int col0, int lane) {
  const int c = col0 + (lane & 15);
  const int kk = k0 + 2 * (lane >> 4);
  v2f b;
  b.x = base[kk * ld + c];
  b.y = base[(kk + 1) * ld + c];
  return b;
}
__device__ __forceinline__ v8f wmma4(v2f a, v2f b, v8f c) {
  return __builtin_amdgcn_wmma_f32_16x16x4_f32(false, a, false, b, (short)0, c, false, false);
}

// ---- Kernel 1: global average pool over H*W per (n,f,c) ----
__global__ void pool_kernel(const float* __restrict__ x, float* __restrict__ pool) {
  __shared__ float red[256];
  const int tid = threadIdx.x;
  const int b = blockIdx.x;  // ((n*F)+f)*C + c
  const float4* p4 = (const float4*)(x + (size_t)b * kHW);
  float s = 0.f;
  for (int i = tid; i < kHW / 4; i += 256) {
    const float4 v = p4[i];
    s += (v.x + v.y) + (v.z + v.w);
  }
  red[tid] = s;
  __syncthreads();
  for (int off = 128; off > 0; off >>= 1) {
    if (tid < off) red[tid] += red[tid + off];
    __syncthreads();
  }
  if (tid == 0) {
    const int n = b / (kF * kC);
    const int f = (b / kC) % kF;
    const int c = b % kC;
    pool[((size_t)n * kC + c) * kF + f] = red[0] * (1.0f / kHW);
  }
}

// ---- Kernel 2: qk linear + per-head softmax attention (tiny) ----
__global__ void attn_kernel(const float* __restrict__ pool,
                            const float* __restrict__ qk_w,
                            const float* __restrict__ qk_b,
                            float* __restrict__ attn) {
  __shared__ float sp[kC * kF];     // pool rows
  __shared__ float sqk[kC * 128];   // qk outputs
  const int tid = threadIdx.x;
  const int n = blockIdx.x;
  for (int i = tid; i < kC * kF; i += 256) sp[i] = pool[(size_t)n * kC * kF + i];
  __syncthreads();
  for (int i = tid; i < kC * 128; i += 256) {
    const int c = i >> 7, j = i & 127;
    float s = qk_b[j];
    const float* w = qk_w + j * kF;
    const float* pp = sp + c * kF;
    for (int f = 0; f < kF; ++f) s += pp[f] * w[f];
    sqk[i] = s;
  }
  __syncthreads();
  if (tid < kHeads * kC) {
    const int h = tid / kC, c = tid % kC;
    const float scale = 0.35355339059327373f;  // n_heads^-0.5
    float logit[kC];
    float mx = -1e30f;
    for (int e = 0; e < kC; ++e) {
      float s = 0.f;
      for (int d = 0; d < 8; ++d)
        s += sqk[c * 128 + h * 8 + d] * sqk[e * 128 + 64 + h * 8 + d];
      s *= scale;
      logit[e] = s;
      mx = fmaxf(mx, s);
    }
    float sum = 0.f;
    for (int e = 0; e < kC; ++e) { const float t = __expf(logit[e] - mx); logit[e] = t; sum += t; }
    const float inv = 1.0f / sum;
    for (int e = 0; e < kC; ++e)
      attn[(((size_t)n * kHeads + h) * kC + c) * kC + e] = logit[e] * inv;
  }
}

// ---- Kernel 3: fused V = v_w@X, Val = attn@V, out = conv_w@[X;Val] + bias ----
__global__ __launch_bounds__(256, 1) void band_main_kernel(
    const float* __restrict__ x, const float* __restrict__ v_w,
    const float* __restrict__ v_b, const float* __restrict__ conv_w,
    const float* __restrict__ conv_b, const float* __restrict__ attn,
    float* __restrict__ out) {
  __shared__ float s_vw[kF * kVWLD];          // 16.6 KB
  __shared__ float s_cw[kF * kCWLD];          // 33.0 KB
  __shared__ float s_at[kHeads * 32 * kATLD]; // 33.8 KB (zero-padded 32x32/head)
  __shared__ float s_xv[kF * kXLD];           // 33.8 KB : X, then reused for V
  __shared__ float s_val[kF * kXLD];          // 33.8 KB
  __shared__ float s_cb[kF];

  const int tid = threadIdx.x;
  const int lane = tid & 31;
  const int wave = tid >> 5;
  const int blocksPerN = kHW / kPixPerBlock;  // 512
  const int n = blockIdx.x / blocksPerN;
  const int pixBase = (blockIdx.x % blocksPerN) * kPixPerBlock;

  // ---- stage weights / attention into LDS ----
  for (int i = tid; i < kF * kF; i += 256)
    s_vw[(i >> 6) * kVWLD + (i & 63)] = v_w[i];
  for (int i = tid; i < kF * 128; i += 256)
    s_cw[(i >> 7) * kCWLD + (i & 127)] = conv_w[i];
  for (int i = tid; i < kHeads * 32 * 32; i += 256) {
    const int h = i >> 10, r = (i >> 5) & 31, e = i & 31;
    const float a = (r < kC && e < kC)
        ? attn[(((size_t)n * kHeads + h) * kC + r) * kC + e] : 0.0f;
    s_at[(h * 32 + r) * kATLD + e] = a;
  }
  if (tid < kF) {  // fold v_b through attention (softmax rows sum to 1)
    float b = conv_b[tid];
    for (int g = 0; g < kF; ++g) b += conv_w[tid * 128 + 64 + g] * v_b[g];
    s_cb[tid] = b;
  }
  __syncthreads();

  const int m = wave & 3;    // 16-row output-feature tile
  const int cg = wave >> 2;  // column-tile group (4 tiles each)
  const int h = wave;        // head handled in phase 2
  const size_t xN = (size_t)n * kF * kC * kHW;

  // ---- hoist all A-fragments (weights, attention) into registers; they are
  //      invariant across column tiles and across all pixel-tile iterations ----
  v2f wv[16], wc1[16], wc2[16];   // v_w, conv_w[:, :64], conv_w[:, 64:]
#pragma unroll
  for (int k = 0; k < 16; ++k) {
    wv[k]  = frag_a(s_vw, kVWLD, m * 16, k * 4, lane);
    wc1[k] = frag_a(s_cw, kCWLD, m * 16, k * 4, lane);
    wc2[k] = frag_a(s_cw, kCWLD, m * 16, 64 + k * 4, lane);
  }
  v2f wat[2][8];                  // attn_h, two c-tiles x eight e-tiles
#pragma unroll
  for (int mt = 0; mt < 2; ++mt)
#pragma unroll
    for (int k = 0; k < 8; ++k)
      wat[mt][k] = frag_a(&s_at[h * 32 * kATLD], kATLD, mt * 16, k * 4, lane);

  for (int it = 0; it < kIters; ++it) {
    const int pix0 = pixBase + it * kP;

    // ---- stage X[f][c*4+p] ----
    for (int i = tid; i < kF * kC; i += 256) {
      const int f = i / kC, c = i % kC;
      const float4 v = *(const float4*)(x + xN + (size_t)(f * kC + c) * kHW + pix0);
      float* d = &s_xv[f * kXLD + c * 4];
      d[0] = v.x; d[1] = v.y; d[2] = v.z; d[3] = v.w;
    }
    if (tid < kF) {  // zero the pad band (cols 124..127) so V pad = 0
      float* d = &s_xv[tid * kXLD + 124];
      d[0] = 0.f; d[1] = 0.f; d[2] = 0.f; d[3] = 0.f;
    }
    __syncthreads();

    // ---- phase 1: V = v_w @ X and O = conv_w1 @ X (WMMA, K=64) ----
    v8f vacc[4], oacc[4];
#pragma unroll
    for (int t = 0; t < 4; ++t) { vacc[t] = {}; oacc[t] = {}; }
#pragma unroll
    for (int t = 0; t < 4; ++t) {
      const int col0 = (cg * 4 + t) * 16;
#pragma unroll
      for (int k = 0; k < 16; ++k) {
        const v2f b = frag_b(s_xv, kXLD, k * 4, col0, lane);
        vacc[t] = wmma4(wv[k],  b, vacc[t]);
        oacc[t] = wmma4(wc1[k], b, oacc[t]);
      }
    }
    __syncthreads();
    // V overwrites the X buffer
#pragma unroll
    for (int t = 0; t < 4; ++t) {
      const int col = (cg * 4 + t) * 16 + (lane & 15);
#pragma unroll
      for (int r = 0; r < 8; ++r) {
        const int g = m * 16 + r + 8 * (lane >> 4);
        s_xv[g * kXLD + col] = vacc[t][r];
      }
    }
    __syncthreads();

    // ---- phase 2: Val_h[c,(d,p)] = attn_h[c,e] @ V_h[e,(d,p)] (one head/wave) ----
    {
      v8f valacc[2][2];
      valacc[0][0] = {}; valacc[0][1] = {}; valacc[1][0] = {}; valacc[1][1] = {};
#pragma unroll
      for (int k = 0; k < 8; ++k) {  // K = e, padded to 32
        v2f bf[2];
#pragma unroll
        for (int nt = 0; nt < 2; ++nt) {
          const int nn = nt * 16 + (lane & 15);
          const int d = nn >> 2, p = nn & 3;
          const int e0 = k * 4 + 2 * (lane >> 4);
          v2f b;
          b.x = s_xv[(8 * h + d) * kXLD + e0 * 4 + p];
          b.y = s_xv[(8 * h + d) * kXLD + (e0 + 1) * 4 + p];
          bf[nt] = b;
        }
#pragma unroll
        for (int mt = 0; mt < 2; ++mt)
#pragma unroll
          for (int nt = 0; nt < 2; ++nt)
            valacc[mt][nt] = wmma4(wat[mt][k], bf[nt], valacc[mt][nt]);
      }
      // scatter Val to s_val[g][c*4+p]
#pragma unroll
      for (int mt = 0; mt < 2; ++mt)
#pragma unroll
        for (int nt = 0; nt < 2; ++nt) {
          const int nn = nt * 16 + (lane & 15);
          const int d = nn >> 2, p = nn & 3, g = 8 * h + d;
#pragma unroll
          for (int r = 0; r < 8; ++r) {
            const int c = mt * 16 + r + 8 * (lane >> 4);
            s_val[g * kXLD + c * 4 + p] = valacc[mt][nt][r];
          }
        }
    }
    __syncthreads();

    // ---- phase 3: O += conv_w2 @ Val ----
#pragma unroll
    for (int t = 0; t < 4; ++t) {
      const int col0 = (cg * 4 + t) * 16;
#pragma unroll
      for (int k = 0; k < 16; ++k) {
        const v2f b = frag_b(s_val, kXLD, k * 4, col0, lane);
        oacc[t] = wmma4(wc2[k], b, oacc[t]);
      }
    }

    // ---- store with folded bias ----
#pragma unroll
    for (int t = 0; t < 4; ++t) {
      const int col = (cg * 4 + t) * 16 + (lane & 15);
      const int c = col >> 2, p = col & 3;
      if (c < kC) {
#pragma unroll
        for (int r = 0; r < 8; ++r) {
          const int o = m * 16 + r + 8 * (lane >> 4);
          out[(((size_t)n * kF + o) * kC + c) * kHW + pix0 + p] = oacc[t][r] + s_cb[o];
        }
      }
    }
    __syncthreads();
  }
}

extern "C" void kernel_launch(void* const* d_in, const int* in_sizes, int n_in,
                              void* d_out, int out_size, void* d_ws, size_t ws_size,
                              hipStream_t stream) {
  (void)in_sizes; (void)n_in; (void)out_size; (void)ws_size;
  const float* x      = (const float*)d_in[0];
  const float* qk_w   = (const float*)d_in[1];
  const float* qk_b   = (const float*)d_in[2];
  const float* v_w    = (const float*)d_in[3];
  const float* v_b    = (const float*)d_in[4];
  const float* conv_w = (const float*)d_in[5];
  const float* conv_b = (const float*)d_in[6];
  float* out  = (float*)d_out;
  float* pool = (float*)d_ws;              // N*C*F = 3968 floats
  float* attn = (float*)d_ws + 4096;       // N*8*31*31 = 15376 floats

  pool_kernel<<<kN * kF * kC, 256, 0, stream>>>(x, pool);
  attn_kernel<<<kN, 256, 0, stream>>>(pool, qk_w, qk_b, attn);
  band_main_kernel<<<kN * (kHW / kPixPerBlock), 256, 0, stream>>>(
      x, v_w, v_b, conv_w, conv_b, attn, out);
}